// SlidingWindowCmn_16226386444943
// MI455X (gfx1250) — compile-verified
//
#include <hip/hip_runtime.h>
#include <stdint.h>

// Problem constants (fixed by the reference harness)
#define T_FRAMES 60000
#define N_FEAT   80      // floats per frame
#define NF4      20      // float4 groups per frame
#define NBATCH   16
#define WIN      600
#define MINWIN   100
#define NCHUNK   15
#define CHUNK    4000    // frames per chunk (multiple of 32)
#define HALO     608     // >= 601, multiple of 32
#define RING     640     // prefix ring entries (span needed <= 632)
#define SLAB     128     // frames per TDM slab (2 KB per slab)
#define NBUF     4       // TDM slab buffers -> 3 DMA ops in flight per wave

typedef float        fx4   __attribute__((ext_vector_type(4)));
typedef unsigned int u32x4 __attribute__((ext_vector_type(4)));
typedef int          i32x8 __attribute__((ext_vector_type(8)));
typedef int          i32x4 __attribute__((ext_vector_type(4)));

// ---- TDM: 2D column gather, tile = [4 floats] x [cnt frames], row stride 80 floats.
// D# per CDNA5 ISA: group0 {count=1, lds_addr, global_addr, type=2},
// group1 {data_size=4B, tensor_dim0=4, tensor_dim1=cnt, tile_dim0=4, tile_dim1=cnt,
//         tensor_dim0_stride=80}. Groups 2/3 zero (2D tensor).
__device__ __forceinline__ void tdm_issue(const void* ldsPtr, const void* gptr, int cnt)
{
    uint32_t lds = (uint32_t)(uintptr_t)ldsPtr;       // LDS byte offset (low 32 bits of flat LDS addr)
    uint64_t ga  = (uint64_t)(uintptr_t)gptr;
    u32x4 g0 = { 1u,                                   // count=1 (valid descriptor)
                 lds,                                  // lds_addr
                 (uint32_t)ga,                         // global_addr[31:0]
                 (uint32_t)(ga >> 32) | (2u << 30) };  // global_addr[56:32] | type=2
    i32x8 g1 = { 0x00020000,          // data_size=4B (2<<16)
                 0x00040000,          // tensor_dim0=4 (low16 << 16)
                 cnt << 16,           // tensor_dim0 hi | tensor_dim1 lo16
                 0x00040000,          // tensor_dim1 hi | tile_dim0=4
                 cnt,                 // tile_dim1=cnt | tile_dim2=0
                 N_FEAT,              // tensor_dim0_stride = 80 elements
                 0, 0 };              // dim1 stride etc. = 0
    i32x4 gz = { 0, 0, 0, 0 };
#if defined(__clang_major__) && (__clang_major__ >= 23)
    i32x8 gz8 = { 0, 0, 0, 0, 0, 0, 0, 0 };
    __builtin_amdgcn_tensor_load_to_lds(g0, g1, gz, gz, gz8, 0);
#else
    __builtin_amdgcn_tensor_load_to_lds(g0, g1, gz, gz, 0);
#endif
}

__device__ __forceinline__ fx4 incl_scan32(fx4 v, unsigned lane)
{
#pragma unroll
    for (int d = 1; d < 32; d <<= 1) {
        fx4 o;
        o.x = __shfl_up(v.x, (unsigned)d);
        o.y = __shfl_up(v.y, (unsigned)d);
        o.z = __shfl_up(v.z, (unsigned)d);
        o.w = __shfl_up(v.w, (unsigned)d);
        if (lane >= (unsigned)d) v += o;
    }
    return v;
}

__device__ __forceinline__ fx4 rsqrt4(fx4 v)
{
    fx4 r;
    r.x = rsqrtf(v.x); r.y = rsqrtf(v.y); r.z = rsqrtf(v.z); r.w = rsqrtf(v.w);
    return r;
}

__global__ __launch_bounds__(32)
void SlidingWindowCmn_16226386444943_kernel(const float* __restrict__ x,
                                            float* __restrict__ out)
{
    const unsigned lane = threadIdx.x;
    const int f4i = blockIdx.x;   // which float4 feature group [0,20)
    const int c   = blockIdx.y;   // chunk along T            [0,15)
    const int b   = blockIdx.z;   // batch                    [0,16)

    const int t0          = c * CHUNK;
    const int tEnd        = t0 + CHUNK;
    const int local_start = (c == 0) ? 0 : (t0 - HALO);
    const int nLocal      = tEnd - local_start;            // 4000 or 4608, multiple of 32
    const int nSlab       = (nLocal + SLAB - 1) / SLAB;    // >= 32

    __shared__ fx4 ringS[RING];         // prefix sums P(j)
    __shared__ fx4 ringQ[RING];         // prefix sums of squares
    __shared__ fx4 slab[NBUF][SLAB];    // TDM ring of slab buffers (frame-major, 16B/frame)

    const fx4* colBase = (const fx4*)x + ((size_t)b * T_FRAMES) * NF4 + f4i;
    fx4*       outV    = (fx4*)out;

    if (lane == 0) { ringS[0] = (fx4)0.0f; ringQ[0] = (fx4)0.0f; }   // P(0) = 0

    // Prologue: prefetch slabs 0..3 (nSlab >= 32 always).
#pragma unroll
    for (int k = 0; k < NBUF; ++k) {
        int ck = nLocal - k * SLAB; ck = ck > SLAB ? SLAB : ck;
        tdm_issue(&slab[k][0], colBase + (size_t)(local_start + k * SLAB) * NF4, ck);
    }

    fx4 carryS = (fx4)0.0f, carryQ = (fx4)0.0f;

    for (int s = 0; s < nSlab; ++s) {
        // Issued so far = min(nSlab, s+4); TDM completes in-order per wave, so
        // allowing min(3, nSlab-1-s) pending guarantees slab s has landed.
        const int rem = nSlab - 1 - s;
        if (rem >= 3)      __builtin_amdgcn_s_wait_tensorcnt(3);
        else if (rem == 2) __builtin_amdgcn_s_wait_tensorcnt(2);
        else if (rem == 1) __builtin_amdgcn_s_wait_tensorcnt(1);
        else               __builtin_amdgcn_s_wait_tensorcnt(0);
        __asm__ __volatile__("" ::: "memory");

        const int buf   = s & (NBUF - 1);
        const int sBase = s * SLAB;
        int cnt = nLocal - sBase; if (cnt > SLAB) cnt = SLAB;

        for (int tt = 0; tt < cnt; tt += 32) {
            const int j  = sBase + tt + (int)lane;   // local frame index
            const fx4 xv = slab[buf][tt + lane];

            fx4 sv = incl_scan32(xv, lane);
            fx4 qv = incl_scan32(xv * xv, lane);
            const fx4 Ps = carryS + sv;              // P(j+1)
            const fx4 Pq = carryQ + qv;
            carryS.x += __shfl(sv.x, 31); carryS.y += __shfl(sv.y, 31);
            carryS.z += __shfl(sv.z, 31); carryS.w += __shfl(sv.w, 31);
            carryQ.x += __shfl(qv.x, 31); carryQ.y += __shfl(qv.y, 31);
            carryQ.z += __shfl(qv.z, 31); carryQ.w += __shfl(qv.w, 31);

            const unsigned wslot = (unsigned)(j + 1) % RING;
            ringS[wslot] = Ps; ringQ[wslot] = Pq;

            const int t = local_start + j;
            bool emit = (t >= t0) && (t < tEnd);
            if (c == 0 && t < 96) emit = false;      // deferred below (window reaches forward)
            if (emit) {
                int we = t + 1;  if (we < MINWIN) we = MINWIN;   // T never clamps (T >> WIN)
                int ws = t - WIN; if (ws < 0) ws = 0;
                const float invn = 1.0f / (float)(we - ws);
                fx4 Swe = Ps, Qwe = Pq;                          // common case we == t+1
                if (we != t + 1) {
                    unsigned es = (unsigned)(we - local_start) % RING;
                    Swe = ringS[es]; Qwe = ringQ[es];
                }
                const unsigned bs = (unsigned)(ws - local_start) % RING;
                const fx4 Sws = ringS[bs], Qws = ringQ[bs];
                const fx4 mean = (Swe - Sws) * invn;
                const fx4 ex2  = (Qwe - Qws) * invn;
                const fx4 r    = rsqrt4(ex2 - mean * mean);
                const fx4 o    = (xv - mean) * r;
                __builtin_nontemporal_store(o, &outV[((size_t)b * T_FRAMES + t) * NF4 + f4i]);
            }

            // Chunk 0, frames [0,96): window is the fixed block [0,100); emit once P(100) exists
            // (this tile wrote P(97..128)); slab buffer 0 still holds x[0..127].
            if (c == 0 && s == 0 && tt == 96) {
                const fx4 S100 = ringS[100];
                const fx4 Q100 = ringQ[100];
                const float invn = 1.0f / 100.0f;
                const fx4 m = S100 * invn;
                const fx4 r = rsqrt4(Q100 * invn - m * m);
#pragma unroll
                for (int g = 0; g < 3; ++g) {
                    const int td = g * 32 + (int)lane;
                    const fx4 xo = slab[0][td];
                    __builtin_nontemporal_store((xo - m) * r,
                        &outV[((size_t)b * T_FRAMES + td) * NF4 + f4i]);
                }
            }
        }

        // Refill this buffer with slab s+4 (its TDM completed and contents consumed).
        if (s + NBUF < nSlab) {
            __asm__ __volatile__("" ::: "memory");
            int cn = nLocal - (s + NBUF) * SLAB; if (cn > SLAB) cn = SLAB;
            tdm_issue(&slab[buf][0],
                      colBase + (size_t)(local_start + (s + NBUF) * SLAB) * NF4, cn);
        }
    }
}

extern "C" void kernel_launch(void* const* d_in, const int* in_sizes, int n_in,
                              void* d_out, int out_size, void* d_ws, size_t ws_size,
                              hipStream_t stream)
{
    (void)in_sizes; (void)n_in; (void)d_ws; (void)ws_size; (void)out_size;
    const float* x = (const float*)d_in[0];
    float* out     = (float*)d_out;
    dim3 grid(NF4, NCHUNK, NBATCH);   // 20 x 15 x 16 = 4800 single-wave workgroups
    SlidingWindowCmn_16226386444943_kernel<<<grid, dim3(32, 1, 1), 0, stream>>>(x, out);
}